// DiagonalKernelAverageLinear_25031069401681
// MI455X (gfx1250) — compile-verified
//
#include <hip/hip_runtime.h>

// MI455X / gfx1250, wave32. One workgroup (512 thr = 16 waves) per batch image.
//
// Roofline: reads 256*512*512*4B = 268 MB once -> ~11.5us floor @ 23.3 TB/s;
// ~2 GFLOP of adds => purely memory-bound. The ring means are prefix-structured
// (not matmul-shaped), so the streaming phase is scalar/shuffle; WMMA is used
// where the operator actually is a matmul: the per-index 4->1 linear epilogue.
//
// Tiling: wave w -> (band = w>>2: rows [128*band, +128), strip = w&3: cols
// [128*strip, +128)); lane owns 4 consecutive columns, loaded as float4
// (global_load_b128). For a row r in band d, its diagonal column strip is d and
// its anti-diagonal strip is 3-d, so "full strips left of the cut" conditions
// are wave-uniform loop invariants, and one fixed wave per band handles the
// masked partial sums at the diagonal / anti-diagonal cuts.

typedef float v2f __attribute__((ext_vector_type(2)));
typedef float v8f __attribute__((ext_vector_type(8)));

#define IMG_H 512
#define IMG_W 512

__device__ __forceinline__ float wave32_sum(float s) {
    s += __shfl_xor(s, 16, 32);
    s += __shfl_xor(s,  8, 32);
    s += __shfl_xor(s,  4, 32);
    s += __shfl_xor(s,  2, 32);
    s += __shfl_xor(s,  1, 32);
    return s;
}

__global__ __launch_bounds__(512)
void ring_linear_kernel(const float* __restrict__ x,
                        const float* __restrict__ wts,   // [512][4][1]
                        const float* __restrict__ bias,  // [512][1]
                        float* __restrict__ out)         // [256][512]
{
    // Row statistics (indexed by row r):
    __shared__ float sT[512];   // T(r)   = sum_c x[r,c]
    __shared__ float sA1[512];  // sum of full 128-col strips left of diag strip
    __shared__ float sA2[512];  // sum of full strips left of anti-diag strip
    __shared__ float sPD[512];  // partial in diag strip: sum_{c in [128*(r>>7), r]}
    __shared__ float sPA[512];  // partial in anti strip: sum_{c in [128*wa, 511-r]}
    __shared__ float sXD[512];  // x[r, r]
    __shared__ float sXA[512];  // x[r, 511-r]
    // Column statistics (indexed by column c):
    __shared__ float sCT[512];  // sum_r x[r,c]
    __shared__ float sCPD[512]; // sum_{r<c}     x[r,c]
    __shared__ float sCPA[512]; // sum_{r<511-c} x[r,c]
    // Features (tl,tr,bl,br order matching reference channel concat):
    __shared__ float sF[512][4];

    const int b = blockIdx.x;
    const int t = threadIdx.x;
    const int w = t >> 5;          // wave id
    const int l = t & 31;          // lane
    const int band  = w >> 2;      // row band: rows [128*band, 128*band+128)
    const int strip = w & 3;       // col strip: cols [128*strip, 128*strip+128)
    const int c0 = 128 * strip + 4 * l;   // first of this lane's 4 columns

    sT[t] = 0.f; sA1[t] = 0.f; sA2[t] = 0.f;
    sCT[t] = 0.f; sCPD[t] = 0.f; sCPA[t] = 0.f;
    __syncthreads();

    const float* xb = x + (size_t)b * (size_t)(IMG_H * IMG_W);

    // Wave-uniform loop invariants (diag strip of every row in this band is
    // `band`, anti strip is `3-band`):
    const bool addA1    = (strip < band);       // strip fully left of diag cut
    const bool addA2    = (strip < 3 - band);   // strip fully left of anti cut
    const bool diagWave = (strip == band);
    const bool antiWave = (strip == 3 - band);

    float ct0 = 0.f, ct1 = 0.f, ct2 = 0.f, ct3 = 0.f;
    float pd0 = 0.f, pd1 = 0.f, pd2 = 0.f, pd3 = 0.f;   // sum_{r<c}
    float pa0 = 0.f, pa1 = 0.f, pa2 = 0.f, pa3 = 0.f;   // sum_{r<511-c}

    const int r0 = 128 * band;
    for (int i = 0; i < 128; ++i) {
        const int r = r0 + i;
        const float4 v =
            *reinterpret_cast<const float4*>(xb + (size_t)r * IMG_W + c0);

        // per-lane column accumulators (global row thresholds)
        ct0 += v.x; ct1 += v.y; ct2 += v.z; ct3 += v.w;
        if (r < c0 + 0) pd0 += v.x;
        if (r < c0 + 1) pd1 += v.y;
        if (r < c0 + 2) pd2 += v.z;
        if (r < c0 + 3) pd3 += v.w;
        if (r < 511 - (c0 + 0)) pa0 += v.x;
        if (r < 511 - (c0 + 1)) pa1 += v.y;
        if (r < 511 - (c0 + 2)) pa2 += v.z;
        if (r < 511 - (c0 + 3)) pa3 += v.w;

        // 128-column strip row sum
        float s = wave32_sum(v.x + v.y + v.z + v.w);
        if (l == 0) {
            atomicAdd(&sT[r], s);                 // ds_add_f32
            if (addA1) atomicAdd(&sA1[r], s);
            if (addA2) atomicAdd(&sA2[r], s);
        }

        if (diagWave) {                 // masked partial up to column r
            const int cut = r & 127;    // include cols with (4l+j) <= cut
            float m = 0.f;
            if (4 * l + 0 <= cut) m += v.x;
            if (4 * l + 1 <= cut) m += v.y;
            if (4 * l + 2 <= cut) m += v.z;
            if (4 * l + 3 <= cut) m += v.w;
            m = wave32_sum(m);
            if (l == 0) sPD[r] = m;
            if (l == (cut >> 2)) {
                const int sub = cut & 3;
                sXD[r] = sub == 0 ? v.x : sub == 1 ? v.y : sub == 2 ? v.z : v.w;
            }
        }
        if (antiWave) {                 // masked partial up to column 511-r
            const int cut = (511 - r) & 127;
            float m = 0.f;
            if (4 * l + 0 <= cut) m += v.x;
            if (4 * l + 1 <= cut) m += v.y;
            if (4 * l + 2 <= cut) m += v.z;
            if (4 * l + 3 <= cut) m += v.w;
            m = wave32_sum(m);
            if (l == 0) sPA[r] = m;
            if (l == (cut >> 2)) {
                const int sub = cut & 3;
                sXA[r] = sub == 0 ? v.x : sub == 1 ? v.y : sub == 2 ? v.z : v.w;
            }
        }
    }

    // fold this band's column partials into the shared column stats
    atomicAdd(&sCT[c0 + 0], ct0); atomicAdd(&sCT[c0 + 1], ct1);
    atomicAdd(&sCT[c0 + 2], ct2); atomicAdd(&sCT[c0 + 3], ct3);
    atomicAdd(&sCPD[c0 + 0], pd0); atomicAdd(&sCPD[c0 + 1], pd1);
    atomicAdd(&sCPD[c0 + 2], pd2); atomicAdd(&sCPD[c0 + 3], pd3);
    atomicAdd(&sCPA[c0 + 0], pa0); atomicAdd(&sCPA[c0 + 1], pa1);
    atomicAdd(&sCPA[c0 + 2], pa2); atomicAdd(&sCPA[c0 + 3], pa3);
    __syncthreads();

    // ---- Phase 2: combine into corner-ring sums, divide by ring size 2m+1 ----
    {
        const int m  = t;
        const int rm = 511 - m;
        float sd_m  = sA1[m]  + sPD[m];    // sum_{c<=m}     x[m,c]
        float sa_m  = sA2[m]  + sPA[m];    // sum_{c<=511-m} x[m,c]
        float sd_rm = sA1[rm] + sPD[rm];   // sum_{c<=rm}    x[rm,c]
        float sa_rm = sA2[rm] + sPA[rm];   // sum_{c<=m}     x[rm,c]

        // tl ring m: row r=m with c<=m, plus col c=m with r<m
        float tl = sd_m + sCPD[m];
        // tr ring m: row r=m with c>=511-m, plus col c=511-m with r<m
        float tr = (sT[m] - sa_m + sXA[m]) + sCPA[rm];
        // bl ring m: row r=511-m with c<=m, plus col c=m with r>511-m
        float bl = sa_rm + (sCT[m] - sCPA[m] - sXA[rm]);
        // br ring m: row r=511-m with c>=511-m, plus col c=511-m with r>511-m
        float br = (sT[rm] - sd_rm + sXD[rm]) + (sCT[rm] - sCPD[rm] - sXD[rm]);

        float inv = 1.0f / (float)(2 * m + 1);
        sF[m][0] = tl * inv;
        sF[m][1] = tr * inv;
        sF[m][2] = bl * inv;
        sF[m][3] = br * inv;
    }
    __syncthreads();

    // ---- Phase 3: per-index linear via WMMA f32 16x16x4, diag extraction ----
    // A[i,k] = feats[S0+i, k]  (16x4),  B[k,j] = W[S0+j, k]  (4x16)
    // D[i,j] = sum_k A[i,k]*B[k,j]  ->  D[i,i] = out[b, S0+i] (pre bias/relu).
    // A layout: lanes 0-15 hold K=0,1 in v[0],v[1]; lanes 16-31 hold K=2,3.
    // B layout: lanes 0-15 hold rows K=0,1; lanes 16-31 rows K=2,3; N = lane&15.
#pragma unroll
    for (int half = 0; half < 2; ++half) {
        const int S0    = 32 * w + 16 * half;
        const int idx16 = l & 15;              // M for A, N for B
        const int kh    = (l >> 4) << 1;       // K pair base (0 or 2)

        v2f A, B;
        A.x = sF[S0 + idx16][kh];
        A.y = sF[S0 + idx16][kh + 1];
        B.x = wts[(S0 + idx16) * 4 + kh];
        B.y = wts[(S0 + idx16) * 4 + kh + 1];

        v8f Cacc = {};
        v8f D = __builtin_amdgcn_wmma_f32_16x16x4_f32(
            /*neg_a=*/false, A, /*neg_b=*/false, B,
            /*c_mod=*/(short)0, Cacc, /*reuse_a=*/false, /*reuse_b=*/false);

        // Diagonal of 16x16 f32 D: VGPR q, lanes 0-15 -> M=q (N=lane),
        // lanes 16-31 -> M=q+8. Lane q holds D[q][q] in D[q]; lane q+24 holds
        // D[q+8][q+8] in D[q].
        float diag = 0.f; int mm = -1;
#pragma unroll
        for (int q = 0; q < 8; ++q) {
            if (l == q)      { diag = D[q]; mm = S0 + q; }
            if (l == q + 24) { diag = D[q]; mm = S0 + q + 8; }
        }
        if (mm >= 0) {
            float o = diag + bias[mm];
            out[(size_t)b * 512 + mm] = o > 0.f ? o : 0.f;
        }
    }
}

extern "C" void kernel_launch(void* const* d_in, const int* in_sizes, int n_in,
                              void* d_out, int out_size, void* d_ws, size_t ws_size,
                              hipStream_t stream) {
    const float* x    = (const float*)d_in[0];  // [256,1,512,512]
    const float* wts  = (const float*)d_in[1];  // [512,4,1]
    const float* bias = (const float*)d_in[2];  // [512,1]
    float* out = (float*)d_out;                 // [256,512]
    ring_linear_kernel<<<256, 512, 0, stream>>>(x, wts, bias, out);
}